// Noise_89910845374637
// MI455X (gfx1250) — compile-verified
//
#include <hip/hip_runtime.h>
#include <hip/hip_bf16.h>

typedef float v2f __attribute__((ext_vector_type(2)));
typedef float v8f __attribute__((ext_vector_type(8)));

#define N_NODES 2048
#define N_GRAPHS 64
#define N_EDGES 16384
#define BATCH 65536
#define META 64
#define TXD 8
#define NOISE 128
#define TSTRIDE 48          // T tile row stride (44 used + pad)
#define CROWS 64            // batch rows per chunk (4 row-tiles of 16)
#define NCHUNK (BATCH / CROWS)

// map extended-K index -> row of emb_W ([IN_DIM=2089, 128], row-major)
// k 0..31  -> trig rows 2049..2080
// k 32     -> chain row 2048
// k 33..40 -> tx rows 2081..2088
// k 41..43 -> zero-padded (T holds 0.0 there; any in-bounds row works)
__device__ __forceinline__ int kext_row(int k) {
    if (k < 32) return 2049 + k;
    if (k == 32) return 2048;
    if (k <= 40) return 2081 + (k - 33);
    return 2048;
}

// ---------------------------------------------------------------------------
// Kernel 1: per-graph GCN (degrees + sym-normalized aggregation) in LDS.
// graph_emb[g, n] = agg[n] * w + b
// ---------------------------------------------------------------------------
__global__ void __launch_bounds__(256) gcn_kernel(const int* __restrict__ edges,
                                                  const float* __restrict__ gw,
                                                  const float* __restrict__ gb,
                                                  float* __restrict__ graph_emb) {
    __shared__ float deg[N_NODES];   // later reused as dinv
    __shared__ float agg[N_NODES];
    const int g = blockIdx.x;
    const int t = threadIdx.x;
    const int* src = edges + (size_t)g * (2 * N_EDGES);
    const int* dst = src + N_EDGES;

    for (int i = t; i < N_NODES; i += 256) deg[i] = 1.0f;   // self-loop deg
    __syncthreads();
    for (int e = t; e < N_EDGES; e += 256)
        atomicAdd(&deg[dst[e]], 1.0f);
    __syncthreads();
    for (int i = t; i < N_NODES; i += 256) {
        float d = deg[i];
        float r = d > 0.0f ? rsqrtf(d) : 0.0f;
        deg[i] = r;                 // deg[] now holds dinv
        agg[i] = r * r;             // self-loop contribution
    }
    __syncthreads();
    for (int e = t; e < N_EDGES; e += 256) {
        int s = src[e], d = dst[e];
        atomicAdd(&agg[d], deg[s] * deg[d]);
    }
    __syncthreads();
    const float w = *gw, b = *gb;
    for (int i = t; i < N_NODES; i += 256)
        graph_emb[(size_t)g * N_NODES + i] = agg[i] * w + b;
}

// ---------------------------------------------------------------------------
// Kernel 2: graph_proj[64,128] = graph_emb[64,2048] @ emb_W[0:2048, 0:128]
// 4x8 = 32 tiles of 16x16, one wave per tile, v_wmma_f32_16x16x4_f32.
// ---------------------------------------------------------------------------
__global__ void __launch_bounds__(256) proj_kernel(const float* __restrict__ ge,
                                                   const float* __restrict__ embW,
                                                   float* __restrict__ gp) {
    const int wave  = (blockIdx.x * blockDim.x + threadIdx.x) >> 5;  // 0..31
    const int lane  = threadIdx.x & 31;
    const int mtile = wave >> 3;
    const int ntile = wave & 7;
    const int lrow  = lane & 15;
    const int khalf = (lane >> 4) * 2;
    const int arow  = mtile * 16 + lrow;
    const int col   = ntile * 16 + lrow;

    v8f c = {};
    for (int kk = 0; kk < N_NODES; kk += 4) {
        const int k0 = kk + khalf;
        v2f a, bm;
        a.x  = ge[(size_t)arow * N_NODES + k0];
        a.y  = ge[(size_t)arow * N_NODES + k0 + 1];
        bm.x = embW[(size_t)k0 * NOISE + col];
        bm.y = embW[(size_t)(k0 + 1) * NOISE + col];
        c = __builtin_amdgcn_wmma_f32_16x16x4_f32(false, a, false, bm,
                                                  (short)0, c, false, false);
    }
    const int rbase = mtile * 16 + (lane >> 4) * 8;
#pragma unroll
    for (int r = 0; r < 8; ++r)
        gp[(size_t)(rbase + r) * NOISE + col] = c[r];
}

// ---------------------------------------------------------------------------
// Kernel 3: batched epilogue, 64 rows per chunk.
// Step 1 (all 8 waves): T[64,0:32] = relu(trigger[64,64] @ trig_W + trig_b)
//                       T[64,32] = chain ; T[64,33:41] = tx ; T[64,41:44] = 0
// Step 2 (each wave, 4 row-tiles sharing reg-resident B frags):
//   out[64,128] = T[64,44] @ embW[kext_rows,128] + gather(graph_proj) + emb_b
// ---------------------------------------------------------------------------
__global__ void __launch_bounds__(256) batch_kernel(const int* __restrict__ graphs,
                                                    const float* __restrict__ chain,
                                                    const float* __restrict__ trig,
                                                    const float* __restrict__ tx,
                                                    const float* __restrict__ trigW,
                                                    const float* __restrict__ trigb,
                                                    const float* __restrict__ embW,
                                                    const float* __restrict__ embb,
                                                    const float* __restrict__ gp,
                                                    float* __restrict__ out) {
    __shared__ __align__(16) float gpl[N_GRAPHS * NOISE];  // 32 KB graph_proj
    __shared__ float T[CROWS * TSTRIDE];                   // 12 KB staging

    const int t     = threadIdx.x;
    const int lane  = t & 31;
    const int w     = t >> 5;            // wave 0..7
    const int lrow  = lane & 15;
    const int khalf = (lane >> 4) * 2;
    const int col   = w * 16 + lrow;     // step-2 output column 0..127
    const int lr0   = (lane >> 4) * 8;   // D-layout row base within tile

    // cooperative b128 load of graph_proj into LDS
    for (int i = t * 4; i < N_GRAPHS * NOISE; i += 256 * 4)
        *(float4*)&gpl[i] = *(const float4*)&gp[i];

    // hoist step-2 B fragments (constant across all chunks): 11 k-steps of K=4
    v2f bf[11];
#pragma unroll
    for (int ks = 0; ks < 11; ++ks) {
        const int k0 = ks * 4 + khalf;
        bf[ks].x = embW[(size_t)kext_row(k0) * NOISE + col];
        bf[ks].y = embW[(size_t)kext_row(k0 + 1) * NOISE + col];
    }

    // step-1 assignment: wave w -> row-tile rt1 = w>>1, col-tile nt1 = w&1
    const int rt1 = w >> 1;
    const int nt1 = w & 1;
    const int c1  = nt1 * 16 + lrow;     // step-1 output column 0..31
    v2f b1[16];
#pragma unroll
    for (int ks = 0; ks < 16; ++ks) {
        const int k0 = ks * 4 + khalf;
        b1[ks].x = trigW[k0 * 32 + c1];
        b1[ks].y = trigW[(k0 + 1) * 32 + c1];
    }
    const float tb = trigb[c1];
    const float eb = embb[col];
    __syncthreads();  // gpl ready

    for (int chunk = blockIdx.x; chunk < NCHUNK; chunk += gridDim.x) {
        const int rowbase = chunk * CROWS;

        // prefetch next chunk's trigger slab (64 rows x 256 B = 16 KB)
        if (chunk + gridDim.x < NCHUNK) {
            const float* p = trig + (size_t)(chunk + gridDim.x) * CROWS * META
                                  + t * 16;
            __builtin_prefetch(p, 0, 3);
        }

        // fill T cols 32..47: chain, tx, zero pad (64x16 slots, 4 per thread)
#pragma unroll
        for (int j = 0; j < 4; ++j) {
            const int slot = t + 256 * j;
            const int fr = slot >> 4;           // 0..63
            const int fc = 32 + (slot & 15);    // 32..47
            float v = 0.0f;
            const int grow = rowbase + fr;
            if (fc == 32)      v = chain[grow];
            else if (fc <= 40) v = tx[(size_t)grow * TXD + (fc - 33)];
            T[fr * TSTRIDE + fc] = v;
        }

        // step 1: every wave computes one 16x16 tile of relu(trig @ trig_W)
        {
            v8f d = {};
            const int arow = rowbase + rt1 * 16 + lrow;
#pragma unroll
            for (int ks = 0; ks < 16; ++ks) {
                const int k0 = ks * 4 + khalf;
                v2f a;
                a.x = trig[(size_t)arow * META + k0];
                a.y = trig[(size_t)arow * META + k0 + 1];
                d = __builtin_amdgcn_wmma_f32_16x16x4_f32(false, a, false, b1[ks],
                                                          (short)0, d, false, false);
            }
#pragma unroll
            for (int r = 0; r < 8; ++r) {
                float v = d[r] + tb;
                T[(rt1 * 16 + lr0 + r) * TSTRIDE + c1] = v > 0.0f ? v : 0.0f;
            }
        }
        __syncthreads();  // T complete

        // step 2: 4 independent accumulator chains share bf[] (K = 44)
        v8f acc0 = {}, acc1 = {}, acc2 = {}, acc3 = {};
#pragma unroll
        for (int ks = 0; ks < 11; ++ks) {
            const int k0 = ks * 4 + khalf;
            v2f a0, a1, a2, a3;
            a0.x = T[(0 * 16 + lrow) * TSTRIDE + k0];
            a0.y = T[(0 * 16 + lrow) * TSTRIDE + k0 + 1];
            a1.x = T[(1 * 16 + lrow) * TSTRIDE + k0];
            a1.y = T[(1 * 16 + lrow) * TSTRIDE + k0 + 1];
            a2.x = T[(2 * 16 + lrow) * TSTRIDE + k0];
            a2.y = T[(2 * 16 + lrow) * TSTRIDE + k0 + 1];
            a3.x = T[(3 * 16 + lrow) * TSTRIDE + k0];
            a3.y = T[(3 * 16 + lrow) * TSTRIDE + k0 + 1];
            acc0 = __builtin_amdgcn_wmma_f32_16x16x4_f32(false, a0, false, bf[ks],
                                                         (short)0, acc0, false, false);
            acc1 = __builtin_amdgcn_wmma_f32_16x16x4_f32(false, a1, false, bf[ks],
                                                         (short)0, acc1, false, false);
            acc2 = __builtin_amdgcn_wmma_f32_16x16x4_f32(false, a2, false, bf[ks],
                                                         (short)0, acc2, false, false);
            acc3 = __builtin_amdgcn_wmma_f32_16x16x4_f32(false, a3, false, bf[ks],
                                                         (short)0, acc3, false, false);
        }

        // epilogue: + graph_proj gather (LDS) + emb_b
#pragma unroll
        for (int rt = 0; rt < 4; ++rt) {
            const v8f& c = (rt == 0) ? acc0 : (rt == 1) ? acc1
                         : (rt == 2) ? acc2 : acc3;
#pragma unroll
            for (int r = 0; r < 8; ++r) {
                const int grow = rowbase + rt * 16 + lr0 + r;
                const int g = graphs[grow];
                out[(size_t)grow * NOISE + col] = c[r] + gpl[g * NOISE + col] + eb;
            }
        }
        __syncthreads();  // protect T before next chunk rewrites it
    }
}

extern "C" void kernel_launch(void* const* d_in, const int* in_sizes, int n_in,
                              void* d_out, int out_size, void* d_ws, size_t ws_size,
                              hipStream_t stream) {
    const int*   graphs = (const int*)d_in[0];
    const float* chain  = (const float*)d_in[1];
    const float* trig   = (const float*)d_in[2];
    const float* tx     = (const float*)d_in[3];
    const int*   edges  = (const int*)d_in[4];
    const float* gw     = (const float*)d_in[5];
    const float* gb     = (const float*)d_in[6];
    const float* trigW  = (const float*)d_in[7];
    const float* trigb  = (const float*)d_in[8];
    const float* embW   = (const float*)d_in[9];
    const float* embb   = (const float*)d_in[10];
    float* out = (float*)d_out;

    // scratch layout: graph_emb [64*2048] then graph_proj [64*128]  (~544 KB)
    float* ge = (float*)d_ws;
    float* gp = ge + (size_t)N_GRAPHS * N_NODES;

    gcn_kernel<<<N_GRAPHS, 256, 0, stream>>>(edges, gw, gb, ge);
    proj_kernel<<<4, 256, 0, stream>>>(ge, embW, gp);
    batch_kernel<<<512, 256, 0, stream>>>(graphs, chain, trig, tx, trigW, trigb,
                                          embW, embb, gp, out);
}